// RelationPool_5583457485090
// MI455X (gfx1250) — compile-verified
//
#include <hip/hip_runtime.h>

// Problem constants (match reference)
#define BB   2
#define NN   512
#define DD   64
#define HH   128
#define OUTN 64
#define NT   32          // N/16 tiles per dimension
#define HS_STRIDE 132    // padded LDS row stride (floats) to avoid bank conflicts

typedef __attribute__((ext_vector_type(2))) float v2f;
typedef __attribute__((ext_vector_type(8))) float v8f;

// ---------------------------------------------------------------------------
// Kernel 1: xa[b,n,h] = x[b,n,:] @ W1[:D, h] + b1[h]   (b1 folded here)
//           xb[b,n,h] = x[b,n,:] @ W1[D:, h]
// grid = B*N blocks, 128 threads (one per h)
// ---------------------------------------------------------------------------
__global__ __launch_bounds__(128)
void k_precompute(const float* __restrict__ x, const float* __restrict__ W1,
                  const float* __restrict__ b1,
                  float* __restrict__ xa, float* __restrict__ xb) {
    const int bn = blockIdx.x;          // 0 .. B*N-1
    const int h  = threadIdx.x;         // 0 .. 127
    __shared__ float xs[DD];
    if (h < DD) xs[h] = x[(size_t)bn * DD + h];
    __syncthreads();
    float sa = b1[h];
    float sb = 0.0f;
#pragma unroll 8
    for (int d = 0; d < DD; ++d) {
        const float xv = xs[d];
        sa = fmaf(xv, W1[d * HH + h], sa);
        sb = fmaf(xv, W1[(DD + d) * HH + h], sb);
    }
    xa[(size_t)bn * HH + h] = sa;
    xb[(size_t)bn * HH + h] = sb;
}

// ---------------------------------------------------------------------------
// Kernel 2: fused pair MLP + pooling of h2.
// One WG handles a 16x16 (i,j) tile: 256 pair-rows, H=128 cols of W2.
// Steady-state per i-row t: build h1 tile in LDS from registers (xa) + LDS
// (xb), preload the whole A tile into 32 v2f regs (one s_wait_dscnt), then
// 32 back-to-back V_WMMA_F32_16X16X4_F32 per wave against the register-
// resident W2 strip. Column sums of relu(h2+b2) stay in registers; each WG
// writes a deterministic 128-float partial (no float atomics).
// ---------------------------------------------------------------------------
__global__ __launch_bounds__(256)
void k_pairs(const float* __restrict__ xa, const float* __restrict__ xb,
             const float* __restrict__ W2, const float* __restrict__ b2,
             float* __restrict__ partials) {
    const int tid   = threadIdx.x;
    const int lane  = tid & 31;
    const int wave  = tid >> 5;        // 0..7 -> 16-col strip of W2
    const int l15   = lane & 15;
    const int khalf = lane >> 4;       // 0: K=0,1  1: K=2,3 (A/B frag halves)

    const int wg   = blockIdx.x;       // b*1024 + tile
    const int b    = wg >> 10;
    const int tile = wg & 1023;
    const int i0   = (tile >> 5) << 4;
    const int j0   = (tile & (NT - 1)) << 4;

    __shared__ float hs[16 * HS_STRIDE];   // h1 tile (16 rows x 128, padded)
    __shared__ float xbs[16 * HS_STRIDE];  // xb tile (16 i-rows x 128, padded)

    // Preload this wave's 16-column strip of W2 as B-fragments.
    // B is 4x16 (KxN): lane l -> col = 16*wave + (l&15); v0/v1 hold
    // K = 4*kk + 2*khalf + {0,1}.
    const int ncol = wave * 16 + l15;
    v2f Bf[32];
#pragma unroll
    for (int kk = 0; kk < 32; ++kk) {
        const int k = kk * 4 + 2 * khalf;
        Bf[kk].x = W2[(k + 0) * HH + ncol];
        Bf[kk].y = W2[(k + 1) * HH + ncol];
    }
    const float bias = b2[ncol];

    const float* __restrict__ xaB = xa + ((size_t)b * NN + j0) * HH; // rows j0..j0+15
    const float* __restrict__ xbB = xb + ((size_t)b * NN + i0) * HH; // rows i0..i0+15

    // One-time staging: xa tile -> registers, xb tile -> LDS.
    // Thread e-slot mapping: idx = e*256+tid -> m = idx>>7, k = idx&127.
    float xareg[8];
#pragma unroll
    for (int e = 0; e < 8; ++e) {
        const int idx = e * 256 + tid;
        const int m = idx >> 7;
        const int k = idx & 127;
        xareg[e] = xaB[m * HH + k];
        xbs[m * HS_STRIDE + k] = xbB[m * HH + k];
    }

    float csum = 0.0f;

    for (int t = 0; t < 16; ++t) {
        __syncthreads();
        // hs[m][k] = relu(xa[j0+m][k] + xb[i0+t][k]); registers + LDS only
#pragma unroll
        for (int e = 0; e < 8; ++e) {
            const int idx = e * 256 + tid;
            const int m = idx >> 7;
            const int k = idx & 127;
            const float v = xareg[e] + xbs[t * HS_STRIDE + k];
            hs[m * HS_STRIDE + k] = v > 0.0f ? v : 0.0f;
        }
        __syncthreads();

        // Preload the whole A tile for this wave's rows: 16 DS loads in
        // flight, then a single wait, then an uninterrupted WMMA chain.
        v2f Af[32];
#pragma unroll
        for (int kk = 0; kk < 32; ++kk) {
            const int k = kk * 4 + 2 * khalf;
            Af[kk].x = hs[l15 * HS_STRIDE + k];
            Af[kk].y = hs[l15 * HS_STRIDE + k + 1];
        }

        v8f acc = {};
#pragma unroll
        for (int kk = 0; kk < 32; ++kk) {
            // D = A(16x4 f32) x B(4x16 f32) + C, exact fp32 on the matrix pipe
            acc = __builtin_amdgcn_wmma_f32_16x16x4_f32(
                false, Af[kk], false, Bf[kk], (short)0, acc, false, false);
        }
        // acc[r]: rows M = r + 8*khalf, col = ncol. bias + relu + row-sum.
#pragma unroll
        for (int r = 0; r < 8; ++r) {
            const float v = acc[r] + bias;
            csum += v > 0.0f ? v : 0.0f;
        }
    }
    // lanes l and l^16 hold the two row-halves of the same column
    csum += __shfl_xor(csum, 16, 32);
    if (lane < 16) partials[(size_t)wg * HH + ncol] = csum;
}

// ---------------------------------------------------------------------------
// Kernel 3: deterministic partial reduction + tiny epilogue.
// pooled = (sum partials) ; rel = pooled@W3 + N*N*b3 ;
// o = relu(rel@V1 + c1) ; out = o@V2 + c2.  One WG per batch.
// ---------------------------------------------------------------------------
__global__ __launch_bounds__(128)
void k_final(const float* __restrict__ partials,
             const float* __restrict__ W3, const float* __restrict__ b3,
             const float* __restrict__ V1, const float* __restrict__ c1,
             const float* __restrict__ V2, const float* __restrict__ c2,
             float* __restrict__ out) {
    const int b = blockIdx.x;
    const int h = threadIdx.x;          // 0..127
    __shared__ float s0[HH];
    __shared__ float s1[HH];

    float s = 0.0f;
    const float* __restrict__ p = partials + (size_t)b * (NT * NT) * HH;
    for (int t = 0; t < NT * NT; ++t) s += p[(size_t)t * HH + h];
    s0[h] = s;                          // pooled h2 column-sums
    __syncthreads();

    float r = (float)NN * (float)NN * b3[h];
#pragma unroll 8
    for (int k = 0; k < HH; ++k) r = fmaf(s0[k], W3[k * HH + h], r);
    __syncthreads();
    s1[h] = r;                          // pooled rel
    __syncthreads();

    float o = c1[h];
#pragma unroll 8
    for (int k = 0; k < HH; ++k) o = fmaf(s1[k], V1[k * HH + h], o);
    o = o > 0.0f ? o : 0.0f;
    __syncthreads();
    s0[h] = o;
    __syncthreads();

    if (h < OUTN) {
        float y = c2[h];
#pragma unroll 8
        for (int k = 0; k < HH; ++k) y = fmaf(s0[k], V2[k * OUTN + h], y);
        out[(size_t)b * OUTN + h] = y;
    }
}

// ---------------------------------------------------------------------------
extern "C" void kernel_launch(void* const* d_in, const int* in_sizes, int n_in,
                              void* d_out, int out_size, void* d_ws, size_t ws_size,
                              hipStream_t stream) {
    (void)in_sizes; (void)n_in; (void)out_size; (void)ws_size;
    const float* x  = (const float*)d_in[0];
    const float* W1 = (const float*)d_in[1];
    const float* b1 = (const float*)d_in[2];
    const float* W2 = (const float*)d_in[3];
    const float* b2 = (const float*)d_in[4];
    const float* W3 = (const float*)d_in[5];
    const float* b3 = (const float*)d_in[6];
    const float* V1 = (const float*)d_in[7];
    const float* c1 = (const float*)d_in[8];
    const float* V2 = (const float*)d_in[9];
    const float* c2 = (const float*)d_in[10];

    float* ws = (float*)d_ws;
    float* xa = ws;                                  // B*N*H floats
    float* xb = xa + (size_t)BB * NN * HH;           // B*N*H floats
    float* partials = xb + (size_t)BB * NN * HH;     // B*1024*H floats (~1 MB)

    k_precompute<<<BB * NN, 128, 0, stream>>>(x, W1, b1, xa, xb);
    k_pairs<<<BB * NT * NT, 256, 0, stream>>>(xa, xb, W2, b2, partials);
    k_final<<<BB, 128, 0, stream>>>(partials, W3, b3, V1, c1, V2, c2, (float*)d_out);
}